// OhemMSELoss2_53584011985659
// MI455X (gfx1250) — compile-verified
//
#include <hip/hip_runtime.h>
#include <stdint.h>

typedef float v2f __attribute__((ext_vector_type(2)));
typedef float v8f __attribute__((ext_vector_type(8)));

#define HI_BINS 32768   // sq >= 0 -> bits[31]=0 -> bits>>16 in [0, 32767]
#define LO_BINS 65536
#define RED_WGS 2048
#define NORM_TERM 8388608.0f   // 8 * 1024 * 1024 = 2^23 (exact power of two)

// ---------------------------------------------------------------------------
// Workspace layout (uint32 words):
//   ws[0..3]                 : ctrl {selHi, kRem, thrBits, pad}
//   ws[4 .. 4+32767]         : hist1 (hi 15 bits)
//   ws[4+32768 .. +65535]    : hist2 (lo 16 bits)
//   then: float partSum[RED_WGS], float partCnt[RED_WGS]
// ---------------------------------------------------------------------------

__global__ void ohem_ws_zero(uint32_t* ws, int nwords) {
    int i = blockIdx.x * blockDim.x + threadIdx.x;
    if (i < nwords) ws[i] = 0u;
}

// Pass 1: LDS-privatized histogram of top 15 bits of sq's bit pattern.
__global__ __launch_bounds__(256) void ohem_hist_hi(const float4* __restrict__ p4,
                                                    const float4* __restrict__ t4,
                                                    uint32_t* __restrict__ gh, int n4) {
    extern __shared__ uint32_t lh[];   // HI_BINS words = 128 KB (dynamic LDS, CDNA5 WGP has 320KB)
    for (int i = threadIdx.x; i < HI_BINS; i += 256) lh[i] = 0u;
    __syncthreads();
    const int stride = gridDim.x * 256;
    for (int i = blockIdx.x * 256 + threadIdx.x; i < n4; i += stride) {
        __builtin_prefetch(&p4[i + stride], 0, 0);   // -> global_prefetch_b8
        __builtin_prefetch(&t4[i + stride], 0, 0);
        float4 p = p4[i], t = t4[i];
        float d0 = p.x - t.x, d1 = p.y - t.y, d2 = p.z - t.z, d3 = p.w - t.w;
        atomicAdd(&lh[__float_as_uint(d0 * d0) >> 16], 1u);
        atomicAdd(&lh[__float_as_uint(d1 * d1) >> 16], 1u);
        atomicAdd(&lh[__float_as_uint(d2 * d2) >> 16], 1u);
        atomicAdd(&lh[__float_as_uint(d3 * d3) >> 16], 1u);
    }
    __syncthreads();
    for (int i = threadIdx.x; i < HI_BINS; i += 256) {
        uint32_t c = lh[i];
        if (c) atomicAdd(&gh[i], c);
    }
}

// Find hi bucket containing the k-th largest value (descending walk).
__global__ __launch_bounds__(1024) void ohem_select_hi(const uint32_t* __restrict__ gh,
                                                       const int* __restrict__ minkept,
                                                       uint32_t* __restrict__ ctrl) {
    __shared__ uint32_t csum[1024];
    const int C = HI_BINS / 1024;  // 32 bins per thread
    uint32_t s = 0;
    const int base = threadIdx.x * C;
    for (int j = 0; j < C; ++j) s += gh[base + j];
    csum[threadIdx.x] = s;
    __syncthreads();
    if (threadIdx.x == 0) {
        const uint32_t k = (uint32_t)minkept[0];
        uint32_t cum = 0;
        int ch = 1023;
        while (ch > 0 && cum + csum[ch] < k) { cum += csum[ch]; --ch; }
        int b = ch * C + C - 1;
        while (b > ch * C && cum + gh[b] < k) { cum += gh[b]; --b; }
        ctrl[0] = (uint32_t)b;   // selected hi bucket
        ctrl[1] = k - cum;       // residual rank inside bucket (>=1)
    }
}

// Pass 2: histogram of low 16 bits among elements whose hi bits match.
// Matching subset is sparse -> plain L2 atomics (p,t mostly L2-resident: 192MB L2).
__global__ __launch_bounds__(256) void ohem_hist_lo(const float4* __restrict__ p4,
                                                    const float4* __restrict__ t4,
                                                    const uint32_t* __restrict__ ctrl,
                                                    uint32_t* __restrict__ gh2, int n4) {
    const uint32_t selHi = ctrl[0];
    const int stride = gridDim.x * 256;
    for (int i = blockIdx.x * 256 + threadIdx.x; i < n4; i += stride) {
        __builtin_prefetch(&p4[i + stride], 0, 0);
        __builtin_prefetch(&t4[i + stride], 0, 0);
        float4 p = p4[i], t = t4[i];
        float d0 = p.x - t.x, d1 = p.y - t.y, d2 = p.z - t.z, d3 = p.w - t.w;
        uint32_t b0 = __float_as_uint(d0 * d0);
        uint32_t b1 = __float_as_uint(d1 * d1);
        uint32_t b2 = __float_as_uint(d2 * d2);
        uint32_t b3 = __float_as_uint(d3 * d3);
        if ((b0 >> 16) == selHi) atomicAdd(&gh2[b0 & 0xFFFFu], 1u);
        if ((b1 >> 16) == selHi) atomicAdd(&gh2[b1 & 0xFFFFu], 1u);
        if ((b2 >> 16) == selHi) atomicAdd(&gh2[b2 & 0xFFFFu], 1u);
        if ((b3 >> 16) == selHi) atomicAdd(&gh2[b3 & 0xFFFFu], 1u);
    }
}

__global__ __launch_bounds__(1024) void ohem_select_lo(const uint32_t* __restrict__ gh2,
                                                       uint32_t* __restrict__ ctrl) {
    __shared__ uint32_t csum[1024];
    const int C = LO_BINS / 1024;  // 64 bins per thread
    uint32_t s = 0;
    const int base = threadIdx.x * C;
    for (int j = 0; j < C; ++j) s += gh2[base + j];
    csum[threadIdx.x] = s;
    __syncthreads();
    if (threadIdx.x == 0) {
        const uint32_t k = ctrl[1];
        uint32_t cum = 0;
        int ch = 1023;
        while (ch > 0 && cum + csum[ch] < k) { cum += csum[ch]; --ch; }
        int b = ch * C + C - 1;
        while (b > ch * C && cum + gh2[b] < k) { cum += gh2[b]; --b; }
        // exact bit pattern of the k-th largest sq value
        ctrl[2] = (ctrl[0] << 16) | (uint32_t)b;
    }
}

// Issue three 16B/lane async copies (p,t,w tiles) into this wave's LDS slot.
__device__ __forceinline__ void async_stage3(const float* p, const float* t, const float* wt,
                                             uint64_t byteOff,
                                             float* ldsP, float* ldsT, float* ldsW) {
    uint32_t lp = (uint32_t)(uintptr_t)ldsP;   // low 32 bits of flat shared addr == LDS offset
    uint32_t lt = (uint32_t)(uintptr_t)ldsT;
    uint32_t lw = (uint32_t)(uintptr_t)ldsW;
    uint64_t gp = (uint64_t)(uintptr_t)p + byteOff;
    uint64_t gt = (uint64_t)(uintptr_t)t + byteOff;
    uint64_t gw = (uint64_t)(uintptr_t)wt + byteOff;
    asm volatile("global_load_async_to_lds_b128 %0, %1, off" :: "v"(lp), "v"(gp) : "memory");
    asm volatile("global_load_async_to_lds_b128 %0, %1, off" :: "v"(lt), "v"(gt) : "memory");
    asm volatile("global_load_async_to_lds_b128 %0, %1, off" :: "v"(lw), "v"(gw) : "memory");
}

// Pass 3: masked weighted sum + count, streamed via double-buffered async-to-LDS.
__global__ __launch_bounds__(256) void ohem_reduce(const float* __restrict__ p,
                                                   const float* __restrict__ t,
                                                   const float* __restrict__ wt,
                                                   const uint32_t* __restrict__ ctrl,
                                                   float* __restrict__ partSum,
                                                   float* __restrict__ partCnt, int n) {
    __shared__ float stage[8][2][3][128];  // [wave][slot][array][elem] = 24 KB
    __shared__ float redS[8], redC[8];
    const uint32_t thr = ctrl[2];
    const int lane = threadIdx.x & 31;
    const int wave = threadIdx.x >> 5;
    const int W    = gridDim.x * 8;              // total waves
    const int nch  = n >> 7;                     // 128-element chunks (N = 2^24 -> exact)
    float sum = 0.0f, cnt = 0.0f;

    int c = blockIdx.x * 8 + wave;
    int slot = 0;
    if (c < nch)
        async_stage3(p, t, wt, ((uint64_t)c * 128 + (uint64_t)lane * 4) * 4,
                     &stage[wave][0][0][lane * 4], &stage[wave][0][1][lane * 4],
                     &stage[wave][0][2][lane * 4]);
    for (; c < nch; c += W) {
        const int cn = c + W;
        if (cn < nch) {
            async_stage3(p, t, wt, ((uint64_t)cn * 128 + (uint64_t)lane * 4) * 4,
                         &stage[wave][slot ^ 1][0][lane * 4], &stage[wave][slot ^ 1][1][lane * 4],
                         &stage[wave][slot ^ 1][2][lane * 4]);
            asm volatile("s_wait_asynccnt 3" ::: "memory");  // oldest 3 (current slot) done
        } else {
            asm volatile("s_wait_asynccnt 0" ::: "memory");
        }
#pragma unroll
        for (int j = 0; j < 4; ++j) {
            const int idx = lane * 4 + j;
            float pv = stage[wave][slot][0][idx];
            float tv = stage[wave][slot][1][idx];
            float wv = stage[wave][slot][2][idx];
            float d = pv - tv;
            float sq = d * d;
            if (__float_as_uint(sq) > thr) {  // strict >, uint order == float order (sq>=0)
                sum += wv * sq;
                cnt += 1.0f;
            }
        }
        slot ^= 1;
    }
    // wave32 tree reduction
    for (int o = 16; o > 0; o >>= 1) {
        sum += __shfl_down(sum, o, 32);
        cnt += __shfl_down(cnt, o, 32);
    }
    if (lane == 0) { redS[wave] = sum; redC[wave] = cnt; }
    __syncthreads();
    if (threadIdx.x == 0) {
        float s = 0.0f, c2 = 0.0f;
        for (int i = 0; i < 8; ++i) { s += redS[i]; c2 += redC[i]; }
        partSum[blockIdx.x] = s;
        partCnt[blockIdx.x] = c2;
    }
}

// Final: one wave (EXEC all ones). Fold G partials into 64 lane-accumulators,
// collapse with v_wmma_f32_16x16x4_f32 (B = ones -> row sums), finish scalar.
__global__ __launch_bounds__(32) void ohem_final(const float* __restrict__ partSum,
                                                 const float* __restrict__ partCnt,
                                                 int G, float* __restrict__ out) {
    const int lane = threadIdx.x;
    v2f aS = {0.0f, 0.0f}, aC = {0.0f, 0.0f};
    for (int i = lane; i < G; i += 64)        { aS.x += partSum[i]; aC.x += partCnt[i]; }
    for (int i = lane + 32; i < G; i += 64)   { aS.y += partSum[i]; aC.y += partCnt[i]; }
    float s, c;
#if __has_builtin(__builtin_amdgcn_wmma_f32_16x16x4_f32)
    v2f ones = {1.0f, 1.0f};
    v8f z = {};
    // A = 16x4 accumulators (2 VGPRs/lane), B = 4x16 ones -> D[m][n] = rowsum(m)
    v8f dS = __builtin_amdgcn_wmma_f32_16x16x4_f32(false, aS, false, ones, (short)0, z, false, false);
    v8f dC = __builtin_amdgcn_wmma_f32_16x16x4_f32(false, aC, false, ones, (short)0, z, false, false);
    s = dS[0] + dS[1] + dS[2] + dS[3] + dS[4] + dS[5] + dS[6] + dS[7]; // lane0: M=0..7, lane16: M=8..15
    c = dC[0] + dC[1] + dC[2] + dC[3] + dC[4] + dC[5] + dC[6] + dC[7];
    s += __shfl_down(s, 16, 32);
    c += __shfl_down(c, 16, 32);
#else
    s = aS.x + aS.y; c = aC.x + aC.y;
    for (int o = 16; o > 0; o >>= 1) { s += __shfl_down(s, o, 32); c += __shfl_down(c, o, 32); }
#endif
    if (lane == 0) out[0] = s / (NORM_TERM * c);  // sum(w*sq)/norm / cnt
}

extern "C" void kernel_launch(void* const* d_in, const int* in_sizes, int n_in,
                              void* d_out, int out_size, void* d_ws, size_t ws_size,
                              hipStream_t stream) {
    const float* p  = (const float*)d_in[0];
    const float* t  = (const float*)d_in[1];
    const float* w  = (const float*)d_in[2];
    const int*   mk = (const int*)d_in[3];
    float* out = (float*)d_out;
    const int n  = in_sizes[0];   // 16,777,216
    const int n4 = n >> 2;

    uint32_t* ws   = (uint32_t*)d_ws;
    uint32_t* ctrl = ws;
    uint32_t* gh1  = ws + 4;
    uint32_t* gh2  = ws + 4 + HI_BINS;
    float* partS   = (float*)(ws + 4 + HI_BINS + LO_BINS);
    float* partC   = partS + RED_WGS;
    const int zwords = 4 + HI_BINS + LO_BINS;

    ohem_ws_zero<<<(zwords + 255) / 256, 256, 0, stream>>>(ws, zwords);
    ohem_hist_hi<<<2048, 256, HI_BINS * sizeof(uint32_t), stream>>>(
        (const float4*)p, (const float4*)t, gh1, n4);
    ohem_select_hi<<<1, 1024, 0, stream>>>(gh1, mk, ctrl);
    ohem_hist_lo<<<2048, 256, 0, stream>>>((const float4*)p, (const float4*)t, ctrl, gh2, n4);
    ohem_select_lo<<<1, 1024, 0, stream>>>(gh2, ctrl);
    ohem_reduce<<<RED_WGS, 256, 0, stream>>>(p, t, w, ctrl, partS, partC, n);
    ohem_final<<<1, 32, 0, stream>>>(partS, partC, RED_WGS, out);
}